// treeNet_31739808318183
// MI455X (gfx1250) — compile-verified
//
#include <hip/hip_runtime.h>

// Problem constants (from reference)
#define N_ROWS      262144
#define D_IN        512
#define H_DIM       100
#define H_PAD       112      // 7 tiles of 16 columns
#define N_NODES     63
#define N_LEAVES    64
#define N_CLASSES   100
#define C_PAD       112      // 7 tiles of 16 classes
#define MAX_DEPTH   6

#define ROWS_PER_BLOCK 128
#define THREADS     256
#define WAVES       8

// ---- LDS layout (floats) ----
// Phase A region: W1^T stored k-pair interleaved: WtP[k/2][n][2]
//   float offset = ((k>>1)*H_PAD + n)*2 + (k&1)
#define WT_SIZE     (D_IN * H_PAD)                 // 57344 floats = 229376 B
// Overlay region (reused after phase A barrier):
#define H_STRIDE    116
#define H_OFF       0
#define H_SIZE      (WAVES * 16 * H_STRIDE)        // 14848
#define WTH_OFF     (H_OFF + H_SIZE)               // WthP[k/2][j][2], j padded to 64
#define WTH_SIZE    (50 * 64 * 2)                  // 6400  (K=100 -> 50 pairs)
#define D_OFF       (WTH_OFF + WTH_SIZE)
#define D_SIZE      (WAVES * 16 * 64)              // 8192
#define MU_STRIDE   66
#define MU_OFF      (D_OFF + D_SIZE)
#define MU_SIZE     (WAVES * 16 * MU_STRIDE)       // 8448
#define PI_OFF      (MU_OFF + MU_SIZE)             // piP[k/2][c][2], c padded to 112
#define PI_SIZE     (32 * C_PAD * 2)               // 7168  (K=64 -> 32 pairs)
// overlay total = 45056 floats < WT_SIZE (57344)  -> shared bytes = WT_SIZE*4

typedef float v2f __attribute__((ext_vector_type(2)));
typedef float v8f __attribute__((ext_vector_type(8)));

__global__ __launch_bounds__(THREADS, 1)
void treenet_fused(const float* __restrict__ x,
                   const float* __restrict__ W1,
                   const float* __restrict__ b1,
                   const float* __restrict__ Wth,
                   const float* __restrict__ bth,
                   const float* __restrict__ pi,
                   float* __restrict__ out)
{
    extern __shared__ float lds[];
    const int tid   = threadIdx.x;
    const int wave  = tid >> 5;          // wave32
    const int lane  = tid & 31;
    const int m     = lane & 15;         // row (A) or column (B/C) index within tile
    const int khalf = lane >> 4;         // 0 -> K+0/K+1 ; 1 -> K+2/K+3 (ISA A/B layout)
    const long rowBase = (long)blockIdx.x * ROWS_PER_BLOCK + (long)wave * 16;

    const v8f vzero = {0.f, 0.f, 0.f, 0.f, 0.f, 0.f, 0.f, 0.f};

    // ------- stage W1^T (zero-padded, k-pair interleaved) into LDS ------------
    for (int i = tid; i < H_DIM * D_IN; i += THREADS) {
        int n = i >> 9;          // W1 row  (0..99)   D_IN = 512
        int k = i & 511;         // W1 col  (0..511)
        lds[((k >> 1) * H_PAD + n) * 2 + (k & 1)] = W1[i];   // coalesced global read
    }
    for (int i = tid; i < D_IN * (H_PAD - H_DIM); i += THREADS) {
        int k = i / (H_PAD - H_DIM);
        int n = H_DIM + i % (H_PAD - H_DIM);
        lds[((k >> 1) * H_PAD + n) * 2 + (k & 1)] = 0.0f;
    }
    __syncthreads();

    // ------- Phase A: h_tile = x_tile @ W1^T  (fp32 WMMA 16x16x4) -------------
    v8f acc[7];
#pragma unroll
    for (int t = 0; t < 7; ++t) acc[t] = vzero;

    const float* xrow = x + (rowBase + m) * D_IN;
#pragma unroll 8
    for (int k0 = 0; k0 < D_IN; k0 += 4) {
        const int ka = k0 + 2 * khalf;               // even
        // A fragment: one aligned b64 per lane: x[m][ka], x[m][ka+1]
        v2f a = *reinterpret_cast<const v2f*>(xrow + ka);
#pragma unroll
        for (int t = 0; t < 7; ++t) {
            // B fragment: one aligned ds_load_b64, conflict-free across all 32 lanes
            v2f b = *reinterpret_cast<const v2f*>(
                &lds[((ka >> 1) * H_PAD + t * 16 + m) * 2]);
            acc[t] = __builtin_amdgcn_wmma_f32_16x16x4_f32(
                false, a, false, b, (short)0, acc[t], false, false);
        }
    }
    __syncthreads();   // all waves done reading W1^T; overlay region goes live

    // ------- Phase B: bias + ReLU -> h in LDS ---------------------------------
    float* hW = lds + H_OFF + wave * 16 * H_STRIDE;
#pragma unroll
    for (int t = 0; t < 7; ++t) {
#pragma unroll
        for (int v = 0; v < 8; ++v) {
            const int rr = v + 8 * khalf;       // C layout: VGPR v -> rows v / v+8
            const int n  = t * 16 + m;
            const float bias = (n < H_DIM) ? b1[n] : 0.0f;
            const float hv = acc[t][v] + bias;
            hW[rr * H_STRIDE + n] = hv > 0.f ? hv : 0.f;
        }
    }
    // cooperative staging of W_theta^T (k-paired, j padded to 64) and pi (k-paired)
    for (int i = tid; i < H_DIM * 64; i += THREADS) {   // 6400
        int k = i >> 6, j = i & 63;
        lds[WTH_OFF + ((k >> 1) * 64 + j) * 2 + (k & 1)] =
            (j < N_NODES) ? Wth[j * H_DIM + k] : 0.f;
    }
    for (int i = tid; i < N_LEAVES * C_PAD; i += THREADS) {  // 7168
        int k = i / C_PAD, c = i % C_PAD;
        lds[PI_OFF + ((k >> 1) * C_PAD + c) * 2 + (k & 1)] =
            (c < N_CLASSES) ? pi[k * N_CLASSES + c] : 0.f;
    }
    __syncthreads();

    // ------- Phase C: d = sigmoid(h @ Wth^T + bth)  (WMMA, K=100) -------------
    v8f acc2[4];
#pragma unroll
    for (int t = 0; t < 4; ++t) acc2[t] = vzero;
#pragma unroll 5
    for (int ks = 0; ks < 25; ++ks) {
        const int ka = 4 * ks + 2 * khalf;           // even, <= 98
        v2f a = *reinterpret_cast<const v2f*>(&hW[m * H_STRIDE + ka]);
#pragma unroll
        for (int t = 0; t < 4; ++t) {
            v2f b = *reinterpret_cast<const v2f*>(
                &lds[WTH_OFF + ((ka >> 1) * 64 + t * 16 + m) * 2]);
            acc2[t] = __builtin_amdgcn_wmma_f32_16x16x4_f32(
                false, a, false, b, (short)0, acc2[t], false, false);
        }
    }
    float* dW = lds + D_OFF + wave * 16 * 64;
#pragma unroll
    for (int t = 0; t < 4; ++t) {
#pragma unroll
        for (int v = 0; v < 8; ++v) {
            const int j  = t * 16 + m;
            const int rr = v + 8 * khalf;
            const float z = acc2[t][v] + ((j < N_NODES) ? bth[j] : 0.f);
            dW[rr * 64 + j] = 1.f / (1.f + __expf(-z));
        }
    }

    // ------- Phase D1: tree routing -> mu (16 x 64 per wave) ------------------
    float* muW = lds + MU_OFF + wave * 16 * MU_STRIDE;
#pragma unroll 4
    for (int p = 0; p < 32; ++p) {
        const int idx  = p * 32 + lane;       // 16 rows * 64 leaves = 1024 pairs
        const int row  = idx >> 6;
        const int leaf = idx & 63;
        float muv = 1.f;
#pragma unroll
        for (int lvl = 0; lvl < MAX_DEPTH; ++lvl) {
            const int node = ((1 << lvl) - 1) + (leaf >> (MAX_DEPTH - lvl));
            const float dv = dW[row * 64 + node];
            const int bit  = (leaf >> (MAX_DEPTH - 1 - lvl)) & 1;
            muv *= bit ? (1.f - dv) : dv;
        }
        muW[row * MU_STRIDE + leaf] = muv;
    }

    // ------- Phase D2: out = mu @ pi  (WMMA, K=64) ----------------------------
    v8f acc3[7];
#pragma unroll
    for (int t = 0; t < 7; ++t) acc3[t] = vzero;
#pragma unroll 4
    for (int ks = 0; ks < 16; ++ks) {
        const int ka = 4 * ks + 2 * khalf;           // even, <= 62
        v2f a = *reinterpret_cast<const v2f*>(&muW[m * MU_STRIDE + ka]);
#pragma unroll
        for (int t = 0; t < 7; ++t) {
            v2f b = *reinterpret_cast<const v2f*>(
                &lds[PI_OFF + ((ka >> 1) * C_PAD + t * 16 + m) * 2]);
            acc3[t] = __builtin_amdgcn_wmma_f32_16x16x4_f32(
                false, a, false, b, (short)0, acc3[t], false, false);
        }
    }
    // predicated store of the 100 valid columns
#pragma unroll
    for (int t = 0; t < 7; ++t) {
#pragma unroll
        for (int v = 0; v < 8; ++v) {
            const int c  = t * 16 + m;
            const int rr = v + 8 * khalf;
            if (c < N_CLASSES)
                out[(rowBase + rr) * N_CLASSES + c] = acc3[t][v];
        }
    }
}

extern "C" void kernel_launch(void* const* d_in, const int* in_sizes, int n_in,
                              void* d_out, int out_size, void* d_ws, size_t ws_size,
                              hipStream_t stream) {
    const float* x   = (const float*)d_in[0];
    const float* W1  = (const float*)d_in[1];
    const float* b1  = (const float*)d_in[2];
    const float* Wth = (const float*)d_in[3];
    const float* bth = (const float*)d_in[4];
    const float* pi  = (const float*)d_in[5];
    float* outp = (float*)d_out;

    const dim3 grid(N_ROWS / ROWS_PER_BLOCK);
    const dim3 block(THREADS);
    const size_t shmem = (size_t)WT_SIZE * sizeof(float);   // 229376 B < 320 KB WGP LDS
    treenet_fused<<<grid, block, shmem, stream>>>(x, W1, b1, Wth, bth, pi, outp);
}